// DFTPSR_40226663694889
// MI455X (gfx1250) — compile-verified
//
#include <hip/hip_runtime.h>
#include <math.h>

typedef __attribute__((ext_vector_type(2))) float v2f;
typedef __attribute__((ext_vector_type(8))) float v8f;

#define NG    160
#define NG2   25600       // 160*160
#define NG3   4096000     // 160^3
#define KZ    81          // rfft length along z
#define NB    2
#define NC    3
#define NPTS  200000
#define TWO_PI_OVER_N 0.039269908169872414f  // 2*pi/160

// ---- workspace layout (float offsets) ----
#define OFF_G    0u                       // real grid [B][C][x][y][z]; later phi [B][x][y][z]
#define SZ_G     24576000u                // NB*NC*NG^3
#define OFF_S1R  24576000u
#define PL_S     12441600u                // NB*NC*NG*NG*KZ
#define OFF_S1I  (OFF_S1R + PL_S)
#define OFF_S2R  (OFF_S1R + 2u*PL_S)
#define OFF_S2I  (OFF_S2R + PL_S)
#define PL_K     4147200u                 // NB*KZ*NG*NG
#define OFF_PKR  OFF_S2R                  // spectral phi_k (reuses S2)
#define OFF_PKI  (OFF_S2R + 1u*PL_K)
#define OFF_PBR  (OFF_S2R + 2u*PL_K)
#define OFF_PBI  (OFF_S2R + 3u*PL_K)
#define OFF_PCR  (OFF_S2R + 4u*PL_K)
#define OFF_PCI  (OFF_S2R + 5u*PL_K)
#define OFF_TC   (OFF_S2R + 2u*PL_S)      // forward cos table 160x160
#define OFF_TS   (OFF_TC + 25600u)        // forward sin table
#define OFF_TCI  (OFF_TS + 25600u)        // cos/160 (inverse)
#define OFF_TSI  (OFF_TCI + 25600u)       // sin/160 (inverse)
#define OFF_WZR  (OFF_TSI + 25600u)       // irfft-z real weights [z][k] ld=81
#define OFF_WZI  (OFF_WZR + 12960u)       // irfft-z imag weights
#define OFF_PART (OFF_WZI + 12960u)       // per-block partial sums
#define NBLK_PTS 782                      // ceil(200000/256)
#define OFF_PARM (OFF_PART + 2048u)       // {mean0, scale0, mean1, scale1}

// ---- CDNA5 WMMA f32 16x16x4 (wave32) ----
__device__ __forceinline__ v8f wmma4(v2f a, v2f b, v8f c) {
  // D = A(16x4) x B(4x16) + C(16x16), fp32, emits v_wmma_f32_16x16x4_f32
  return __builtin_amdgcn_wmma_f32_16x16x4_f32(false, a, false, b, (short)0, c,
                                               false, false);
}

__device__ __forceinline__ int wrap160(int i) { i %= NG; return i < 0 ? i + NG : i; }

// ------------------------------------------------------------------
// Twiddle tables. Angles computed mod N so sinf/cosf stay accurate.
__global__ void k_init_tables(float* __restrict__ ws) {
  int i = blockIdx.x * blockDim.x + threadIdx.x;
  if (i < 25600) {
    int a = i / NG, b = i % NG;
    int m = (a * b) % NG;
    float ang = (float)m * TWO_PI_OVER_N;
    float c = __cosf(ang), s = __sinf(ang);
    ws[OFF_TC + i]  = c;
    ws[OFF_TS + i]  = s;
    ws[OFF_TCI + i] = c * (1.0f / NG);
    ws[OFF_TSI + i] = s * (1.0f / NG);
  }
  if (i < 12960) {  // irfft-z weights: out[z] = sum_k a_k(cos*Re - sin*Im)/N
    int z = i / KZ, k = i % KZ;
    float alpha = (k == 0 || k == KZ - 1) ? 1.0f : 2.0f;
    int m = (z * k) % NG;
    float ang = (float)m * TWO_PI_OVER_N;
    ws[OFF_WZR + i] =  alpha * __cosf(ang) * (1.0f / NG);
    ws[OFF_WZI + i] = -alpha * __sinf(ang) * (1.0f / NG);
  }
}

__global__ void k_zero(float* __restrict__ ws, unsigned total) {
  unsigned i = blockIdx.x * blockDim.x + threadIdx.x;
  if (i < SZ_G)            ws[OFF_G + i] = 0.0f;
  else if (i < total)      ws[OFF_PART + (i - SZ_G)] = 0.0f;
}

// ------------------------------------------------------------------
// Trilinear scatter-add of normals into G[b][c][x][y][z]
__global__ void k_scatter(const float* __restrict__ v_r,
                          const float* __restrict__ n_r,
                          float* __restrict__ ws) {
  int i = blockIdx.x * blockDim.x + threadIdx.x;
  if (i >= NB * NPTS) return;
  int b = i / NPTS;
  const float* p  = v_r + (size_t)i * 3;
  const float* nv = n_r + (size_t)i * 3;
  float fx[3]; int i0[3], i1[3];
#pragma unroll
  for (int d = 0; d < 3; ++d) {
    float x = p[d] * NG;
    float f = floorf(x);
    fx[d] = x - f;
    i0[d] = wrap160((int)f);
    i1[d] = wrap160(i0[d] + 1);
  }
  float* G = ws + OFF_G;
#pragma unroll
  for (int c = 0; c < 8; ++c) {
    int bx = c & 1, by = (c >> 1) & 1, bz = (c >> 2) & 1;
    int ix = bx ? i1[0] : i0[0];
    int iy = by ? i1[1] : i0[1];
    int iz = bz ? i1[2] : i0[2];
    float w = (bx ? fx[0] : 1.0f - fx[0]) * (by ? fx[1] : 1.0f - fx[1]) *
              (bz ? fx[2] : 1.0f - fx[2]);
    int sp = ix * NG2 + iy * NG + iz;
#pragma unroll
    for (int ch = 0; ch < NC; ++ch)
      atomicAdd(&G[(size_t)(b * NC + ch) * NG3 + sp], nv[ch] * w);
  }
}

// ------------------------------------------------------------------
// Pass 1: rfft along z.  Per slab (b,c,x): OUT[y][kz] = sum_z v[y][z] e^{-i2pi kz z/N}
// GEMM: M=y (10 tiles), N=kz (6 tiles), K=z.  A=data, B=twiddles.
__global__ void k_fft_z(float* __restrict__ ws) {
  int wid = (blockIdx.x * blockDim.x + threadIdx.x) >> 5;
  const int TILES = NB * NC * NG * 60;
  if (wid >= TILES) return;
  int slab = wid / 60, t = wid % 60;
  int mt = t / 6, nt = t % 6;
  int lane = threadIdx.x & 31, hl = lane >> 4, m = lane & 15;
  const float* in = ws + OFF_G + (size_t)slab * NG2;
  const float* Tc = ws + OFF_TC;
  const float* Ts = ws + OFF_TS;
  float* oRe = ws + OFF_S1R + (size_t)slab * (NG * KZ);
  float* oIm = ws + OFF_S1I + (size_t)slab * (NG * KZ);
  int yb = mt * 16, kzc = nt * 16 + m;  // kzc may exceed 80; table reads stay in-range
  v8f aR = {}, aI = {};
  for (int z = 0; z < NG; z += 4) {
    v2f av, bc, bs;
    av.x = in[(yb + m) * NG + z + 2 * hl];
    av.y = in[(yb + m) * NG + z + 2 * hl + 1];
    bc.x = Tc[kzc * NG + z + hl];       bc.y = Tc[kzc * NG + z + 2 + hl];
    bs.x = -Ts[kzc * NG + z + hl];      bs.y = -Ts[kzc * NG + z + 2 + hl];
    aR = wmma4(av, bc, aR);
    aI = wmma4(av, bs, aI);
  }
  if (kzc < KZ) {
#pragma unroll
    for (int r = 0; r < 8; ++r) {
      int y = yb + r + 8 * hl;
      oRe[y * KZ + kzc] = aR[r];
      oIm[y * KZ + kzc] = aI[r];
    }
  }
}

// Pass 2: full FFT along y. Per slab (b,c,x): IN [y][kz] -> OUT [kz][ky]
// GEMM: M=kz (6 tiles), N=ky (10 tiles), K=y.
__global__ void k_fft_y(float* __restrict__ ws) {
  int wid = (blockIdx.x * blockDim.x + threadIdx.x) >> 5;
  const int TILES = NB * NC * NG * 60;
  if (wid >= TILES) return;
  int slab = wid / 60, t = wid % 60;
  int mt = t / 10, nt = t % 10;
  int lane = threadIdx.x & 31, hl = lane >> 4, m = lane & 15;
  const float* iRe = ws + OFF_S1R + (size_t)slab * (NG * KZ);
  const float* iIm = ws + OFF_S1I + (size_t)slab * (NG * KZ);
  const float* Tc = ws + OFF_TC;
  const float* Ts = ws + OFF_TS;
  float* oRe = ws + OFF_S2R + (size_t)slab * (NG * KZ);
  float* oIm = ws + OFF_S2I + (size_t)slab * (NG * KZ);
  int kzb = mt * 16, kyc = nt * 16 + m;
  int akz = kzb + m; if (akz > KZ - 1) akz = KZ - 1;  // clamp padded A rows
  v8f aR = {}, aI = {};
  for (int y = 0; y < NG; y += 4) {
    v2f ar, ai, bc, bs, bns;
    ar.x = iRe[(y + 2 * hl) * KZ + akz];      ar.y = iRe[(y + 2 * hl + 1) * KZ + akz];
    ai.x = iIm[(y + 2 * hl) * KZ + akz];      ai.y = iIm[(y + 2 * hl + 1) * KZ + akz];
    bc.x = Tc[kyc * NG + y + hl];             bc.y = Tc[kyc * NG + y + 2 + hl];
    bs.x = Ts[kyc * NG + y + hl];             bs.y = Ts[kyc * NG + y + 2 + hl];
    bns.x = -bs.x;                            bns.y = -bs.y;
    aR = wmma4(ar, bc, aR);  aR = wmma4(ai, bs, aR);    // Re = C*Re + S*Im
    aI = wmma4(ai, bc, aI);  aI = wmma4(ar, bns, aI);   // Im = C*Im - S*Re
  }
#pragma unroll
  for (int r = 0; r < 8; ++r) {
    int kz = kzb + r + 8 * hl;
    if (kz < KZ) {
      oRe[kz * NG + kyc] = aR[r];
      oIm[kz * NG + kyc] = aI[r];
    }
  }
}

// Pass 3: full FFT along x. Per (b,c): IN [x][kz*160+ky] -> OUT [kz*160+ky][kx]
// GEMM: M=l (810 tiles, l=kz*160+ky), N=kx (10 tiles), K=x.
__global__ void k_fft_x(float* __restrict__ ws) {
  int wid = (blockIdx.x * blockDim.x + threadIdx.x) >> 5;
  const int TILES = NB * NC * 8100;
  if (wid >= TILES) return;
  int bc = wid / 8100, t = wid % 8100;
  int mt = t / 10, nt = t % 10;
  int lane = threadIdx.x & 31, hl = lane >> 4, m = lane & 15;
  const float* iRe = ws + OFF_S2R + (size_t)bc * 2073600;
  const float* iIm = ws + OFF_S2I + (size_t)bc * 2073600;
  const float* Tc = ws + OFF_TC;
  const float* Ts = ws + OFF_TS;
  float* oRe = ws + OFF_S1R + (size_t)bc * 2073600;
  float* oIm = ws + OFF_S1I + (size_t)bc * 2073600;
  int lb = mt * 16, kxc = nt * 16 + m;
  v8f aR = {}, aI = {};
  for (int x = 0; x < NG; x += 4) {
    v2f ar, ai, bcv, bs, bns;
    ar.x = iRe[(x + 2 * hl) * 12960 + lb + m];  ar.y = iRe[(x + 2 * hl + 1) * 12960 + lb + m];
    ai.x = iIm[(x + 2 * hl) * 12960 + lb + m];  ai.y = iIm[(x + 2 * hl + 1) * 12960 + lb + m];
    bcv.x = Tc[kxc * NG + x + hl];              bcv.y = Tc[kxc * NG + x + 2 + hl];
    bs.x  = Ts[kxc * NG + x + hl];              bs.y  = Ts[kxc * NG + x + 2 + hl];
    bns.x = -bs.x;                              bns.y = -bs.y;
    aR = wmma4(ar, bcv, aR);  aR = wmma4(ai, bs, aR);
    aI = wmma4(ai, bcv, aI);  aI = wmma4(ar, bns, aI);
  }
#pragma unroll
  for (int r = 0; r < 8; ++r) {
    int l = lb + r + 8 * hl;
    oRe[l * NG + kxc] = aR[r];
    oIm[l * NG + kxc] = aI[r];
  }
}

// ------------------------------------------------------------------
// Spectral: f_k = i*sum_c s_c v_c ; phi_k = ilap * f_k
__global__ void k_spectral(float* __restrict__ ws) {
  int i = blockIdx.x * blockDim.x + threadIdx.x;
  if (i >= NB * 2073600) return;
  int b = i / 2073600, r = i - b * 2073600;
  int kz = r / NG2, rem = r - kz * NG2;
  int ky = rem / NG, kx = rem - ky * NG;
  float sx = __sinf(kx * TWO_PI_OVER_N) * NG, cx = __cosf(kx * TWO_PI_OVER_N);
  float sy = __sinf(ky * TWO_PI_OVER_N) * NG, cy = __cosf(ky * TWO_PI_OVER_N);
  float sz = __sinf(kz * TWO_PI_OVER_N) * NG, cz = __cosf(kz * TWO_PI_OVER_N);
  float den = 2.0f * (float)NG2 * ((cx - 1.0f) + (cy - 1.0f) + (cz - 1.0f));
  float ilap = (r == 0) ? 0.0f : 1.0f / den;
  float s3[3] = {sx, sy, sz};
  float fre = 0.0f, fim = 0.0f;
#pragma unroll
  for (int c = 0; c < NC; ++c) {
    float vr = ws[OFF_S1R + (size_t)(b * NC + c) * 2073600 + r];
    float vi = ws[OFF_S1I + (size_t)(b * NC + c) * 2073600 + r];
    fre -= s3[c] * vi;   // Re(i*s*v) = -s*Im
    fim += s3[c] * vr;   // Im(i*s*v) =  s*Re
  }
  ws[OFF_PKR + i] = ilap * fre;
  ws[OFF_PKI + i] = ilap * fim;
}

// ------------------------------------------------------------------
// Inverse FFT along x (scale 1/N folded into Tci/Tsi). Per slab (b,kz):
// IN [ky][kx] -> OUT [ky][x].  M=ky, N=x, K=kx.
__global__ void k_ifft_x(float* __restrict__ ws) {
  int wid = (blockIdx.x * blockDim.x + threadIdx.x) >> 5;
  const int TILES = NB * KZ * 100;
  if (wid >= TILES) return;
  int slab = wid / 100, t = wid % 100;
  int mt = t / 10, nt = t % 10;
  int lane = threadIdx.x & 31, hl = lane >> 4, m = lane & 15;
  const float* iRe = ws + OFF_PKR + (size_t)slab * NG2;
  const float* iIm = ws + OFF_PKI + (size_t)slab * NG2;
  const float* Tci = ws + OFF_TCI;
  const float* Tsi = ws + OFF_TSI;
  float* oRe = ws + OFF_PBR + (size_t)slab * NG2;
  float* oIm = ws + OFF_PBI + (size_t)slab * NG2;
  int kyb = mt * 16, xc = nt * 16 + m;
  v8f aR = {}, aI = {};
  for (int kx = 0; kx < NG; kx += 4) {
    v2f ar, ai, bc, bs, bns;
    ar.x = iRe[(kyb + m) * NG + kx + 2 * hl];  ar.y = iRe[(kyb + m) * NG + kx + 2 * hl + 1];
    ai.x = iIm[(kyb + m) * NG + kx + 2 * hl];  ai.y = iIm[(kyb + m) * NG + kx + 2 * hl + 1];
    bc.x = Tci[xc * NG + kx + hl];             bc.y = Tci[xc * NG + kx + 2 + hl];
    bs.x = Tsi[xc * NG + kx + hl];             bs.y = Tsi[xc * NG + kx + 2 + hl];
    bns.x = -bs.x;                             bns.y = -bs.y;
    aR = wmma4(ar, bc, aR);  aR = wmma4(ai, bns, aR);  // Re = C*Re - S*Im
    aI = wmma4(ai, bc, aI);  aI = wmma4(ar, bs, aI);   // Im = C*Im + S*Re
  }
#pragma unroll
  for (int r = 0; r < 8; ++r) {
    int ky = kyb + r + 8 * hl;
    oRe[ky * NG + xc] = aR[r];
    oIm[ky * NG + xc] = aI[r];
  }
}

// Inverse FFT along y. Per slab (b,kz): IN [ky][x] -> OUT [x][y]. M=x, N=y, K=ky.
__global__ void k_ifft_y(float* __restrict__ ws) {
  int wid = (blockIdx.x * blockDim.x + threadIdx.x) >> 5;
  const int TILES = NB * KZ * 100;
  if (wid >= TILES) return;
  int slab = wid / 100, t = wid % 100;
  int mt = t / 10, nt = t % 10;
  int lane = threadIdx.x & 31, hl = lane >> 4, m = lane & 15;
  const float* iRe = ws + OFF_PBR + (size_t)slab * NG2;
  const float* iIm = ws + OFF_PBI + (size_t)slab * NG2;
  const float* Tci = ws + OFF_TCI;
  const float* Tsi = ws + OFF_TSI;
  float* oRe = ws + OFF_PCR + (size_t)slab * NG2;
  float* oIm = ws + OFF_PCI + (size_t)slab * NG2;
  int xb = mt * 16, yc = nt * 16 + m;
  v8f aR = {}, aI = {};
  for (int ky = 0; ky < NG; ky += 4) {
    v2f ar, ai, bc, bs, bns;
    ar.x = iRe[(ky + 2 * hl) * NG + xb + m];   ar.y = iRe[(ky + 2 * hl + 1) * NG + xb + m];
    ai.x = iIm[(ky + 2 * hl) * NG + xb + m];   ai.y = iIm[(ky + 2 * hl + 1) * NG + xb + m];
    bc.x = Tci[yc * NG + ky + hl];             bc.y = Tci[yc * NG + ky + 2 + hl];
    bs.x = Tsi[yc * NG + ky + hl];             bs.y = Tsi[yc * NG + ky + 2 + hl];
    bns.x = -bs.x;                             bns.y = -bs.y;
    aR = wmma4(ar, bc, aR);  aR = wmma4(ai, bns, aR);
    aI = wmma4(ai, bc, aI);  aI = wmma4(ar, bs, aI);
  }
#pragma unroll
  for (int r = 0; r < 8; ++r) {
    int x = xb + r + 8 * hl;
    oRe[x * NG + yc] = aR[r];
    oIm[x * NG + yc] = aI[r];
  }
}

// Inverse rfft along z (real output). Per b: IN [kz][x*160+y] -> phi [x*160+y][z]
// M=l (1600 tiles), N=z (10 tiles), K=kz (81, guarded).
__global__ void k_ifft_z(float* __restrict__ ws) {
  int wid = (blockIdx.x * blockDim.x + threadIdx.x) >> 5;
  const int TILES = NB * 16000;
  if (wid >= TILES) return;
  int b = wid / 16000, t = wid % 16000;
  int mt = t / 10, nt = t % 10;
  int lane = threadIdx.x & 31, hl = lane >> 4, m = lane & 15;
  const float* iRe = ws + OFF_PCR + (size_t)b * 2073600;
  const float* iIm = ws + OFF_PCI + (size_t)b * 2073600;
  const float* Wzr = ws + OFF_WZR;
  const float* Wzi = ws + OFF_WZI;
  float* phi = ws + OFF_G + (size_t)b * NG3;
  int lb = mt * 16, zc = nt * 16 + m;
  v8f acc = {};
  for (int k0 = 0; k0 < KZ; k0 += 4) {
    int ka = k0 + 2 * hl, kb = ka + 1;
    int k1 = k0 + hl, k2 = k0 + 2 + hl;
    v2f ar, ai, br, bi;
    ar.x = (ka < KZ) ? iRe[ka * NG2 + lb + m] : 0.0f;
    ar.y = (kb < KZ) ? iRe[kb * NG2 + lb + m] : 0.0f;
    ai.x = (ka < KZ) ? iIm[ka * NG2 + lb + m] : 0.0f;
    ai.y = (kb < KZ) ? iIm[kb * NG2 + lb + m] : 0.0f;
    br.x = (k1 < KZ) ? Wzr[zc * KZ + k1] : 0.0f;
    br.y = (k2 < KZ) ? Wzr[zc * KZ + k2] : 0.0f;
    bi.x = (k1 < KZ) ? Wzi[zc * KZ + k1] : 0.0f;
    bi.y = (k2 < KZ) ? Wzi[zc * KZ + k2] : 0.0f;
    acc = wmma4(ar, br, acc);
    acc = wmma4(ai, bi, acc);
  }
#pragma unroll
  for (int r = 0; r < 8; ++r) {
    int l = lb + r + 8 * hl;
    phi[l * NG + zc] = acc[r];
  }
}

// ------------------------------------------------------------------
// Trilinear gather of phi at sample points; block partial sums (stage 1)
__global__ void k_interp(const float* __restrict__ v_r, float* __restrict__ ws) {
  __shared__ float sh[256];
  int b = blockIdx.y;
  int n = blockIdx.x * 256 + threadIdx.x;
  float val = 0.0f;
  if (n < NPTS) {
    const float* p = v_r + ((size_t)b * NPTS + n) * 3;
    const float* phi = ws + OFF_G + (size_t)b * NG3;
    float fx[3]; int i0[3], i1[3];
#pragma unroll
    for (int d = 0; d < 3; ++d) {
      float x = p[d] * NG;
      float f = floorf(x);
      fx[d] = x - f;
      i0[d] = wrap160((int)f);
      i1[d] = wrap160(i0[d] + 1);
    }
#pragma unroll
    for (int c = 0; c < 8; ++c) {
      int bx = c & 1, by = (c >> 1) & 1, bz = (c >> 2) & 1;
      float w = (bx ? fx[0] : 1.0f - fx[0]) * (by ? fx[1] : 1.0f - fx[1]) *
                (bz ? fx[2] : 1.0f - fx[2]);
      val += w * phi[(bx ? i1[0] : i0[0]) * NG2 + (by ? i1[1] : i0[1]) * NG +
                     (bz ? i1[2] : i0[2])];
    }
  }
  sh[threadIdx.x] = val;
  __syncthreads();
  for (int s = 128; s > 0; s >>= 1) {
    if (threadIdx.x < s) sh[threadIdx.x] += sh[threadIdx.x + s];
    __syncthreads();
  }
  if (threadIdx.x == 0) ws[OFF_PART + b * NBLK_PTS + blockIdx.x] = sh[0];
}

// Stage 2: deterministic reduction -> mean, scale per batch
__global__ void k_params(float* __restrict__ ws) {
  __shared__ float sh[256];
  int tid = threadIdx.x;
  for (int b = 0; b < NB; ++b) {
    float s = 0.0f;
    for (int i = tid; i < NBLK_PTS; i += 256) s += ws[OFF_PART + b * NBLK_PTS + i];
    sh[tid] = s;
    __syncthreads();
    for (int st = 128; st > 0; st >>= 1) {
      if (tid < st) sh[tid] += sh[tid + st];
      __syncthreads();
    }
    if (tid == 0) {
      float mean = sh[0] / (float)NPTS;
      float p0 = ws[OFF_G + (size_t)b * NG3];
      ws[OFF_PARM + 2 * b]     = mean;
      ws[OFF_PARM + 2 * b + 1] = 0.5f / fabsf(p0 - mean);
    }
    __syncthreads();
  }
}

__global__ void k_final(const float* __restrict__ ws, float* __restrict__ out) {
  int i = blockIdx.x * blockDim.x + threadIdx.x;
  if (i >= NB * NG3) return;
  int b = i / NG3;
  out[i] = (ws[OFF_G + i] - ws[OFF_PARM + 2 * b]) * ws[OFF_PARM + 2 * b + 1];
}

// ------------------------------------------------------------------
extern "C" void kernel_launch(void* const* d_in, const int* in_sizes, int n_in,
                              void* d_out, int out_size, void* d_ws, size_t ws_size,
                              hipStream_t stream) {
  (void)in_sizes; (void)n_in; (void)out_size; (void)ws_size;
  const float* v_r = (const float*)d_in[0];
  const float* n_r = (const float*)d_in[1];
  float* ws  = (float*)d_ws;
  float* out = (float*)d_out;

  k_init_tables<<<100, 256, 0, stream>>>(ws);
  unsigned ztotal = SZ_G + 2056u;
  k_zero<<<(ztotal + 255u) / 256u, 256, 0, stream>>>(ws, ztotal);
  k_scatter<<<(NB * NPTS + 255) / 256, 256, 0, stream>>>(v_r, n_r, ws);

  k_fft_z<<<7200, 256, 0, stream>>>(ws);      // 57600 tiles, 8 waves/block
  k_fft_y<<<7200, 256, 0, stream>>>(ws);
  k_fft_x<<<6075, 256, 0, stream>>>(ws);      // 48600 tiles
  k_spectral<<<16200, 256, 0, stream>>>(ws);
  k_ifft_x<<<2025, 256, 0, stream>>>(ws);     // 16200 tiles
  k_ifft_y<<<2025, 256, 0, stream>>>(ws);
  k_ifft_z<<<4000, 256, 0, stream>>>(ws);     // 32000 tiles

  dim3 gi(NBLK_PTS, NB);
  k_interp<<<gi, 256, 0, stream>>>(v_r, ws);
  k_params<<<1, 256, 0, stream>>>(ws);
  k_final<<<(NB * NG3 + 255) / 256, 256, 0, stream>>>(ws, out);
}